// TTM_25615184953815
// MI455X (gfx1250) — compile-verified
//
#include <hip/hip_runtime.h>
#include <hip/hip_bf16.h>
#include <math.h>

// ---------------------------------------------------------------------------
// Dims (fixed by the reference)
// ---------------------------------------------------------------------------
#define AVGF 32
#define MDIM 2048          // M = C1*C2
#define TDIM 2048
#define SEG  64
#define HA   16
#define DH   128
#define EPSL 1e-5f

typedef float v2f __attribute__((ext_vector_type(2)));
typedef float v4f __attribute__((ext_vector_type(4)));
typedef float v8f __attribute__((ext_vector_type(8)));
typedef unsigned int v4u __attribute__((ext_vector_type(4)));
typedef unsigned int v8u __attribute__((ext_vector_type(8)));

enum EpiMode { EPI_NONE = 0, EPI_SINBIAS = 1, EPI_RES = 2, EPI_BIAS_GELU = 3, EPI_BIAS = 4 };

// ---------------------------------------------------------------------------
// Epilogue helper
// ---------------------------------------------------------------------------
__device__ __forceinline__ float epi_apply(float x, int row, int col, int N,
                                           int epi, const float* __restrict__ extra) {
    switch (epi) {
        case EPI_SINBIAS: {
            // bias[i, m]: ang = i * 10000^{-(m & ~1)/1024}; even col -> sin, odd -> cos
            float jp  = (float)(col & ~1);
            float ang = (float)row * __expf(-jp * (9.210340371976184f / 1024.0f));
            return x + ((col & 1) ? __cosf(ang) : __sinf(ang));
        }
        case EPI_RES:
            return x + extra[(size_t)row * N + col];
        case EPI_BIAS_GELU: {
            float t = x + extra[col];
            return 0.5f * t * (1.0f + erff(t * 0.70710678118654752f));
        }
        case EPI_BIAS:
            return x + extra[col];
        default:
            return x;
    }
}

// ---------------------------------------------------------------------------
// TDM: issue one tensor_load_to_lds moving a 16-row x 32-f32 tile (row stride
// K elems in memory) into LDS with a HW-padded 36-float LDS row stride
// (pad 4 DWORDs after every 32 DWORDs).  Wave-uniform descriptor in SGPRs,
// 2-group form (VADDR2/3 = NULL, tile is 2-D).
// ---------------------------------------------------------------------------
__device__ __forceinline__ void tdm_load_tile16x32(const float* gptr,
                                                   unsigned lds_byte_addr,
                                                   unsigned K_elems) {
    unsigned long long ga = (unsigned long long)(uintptr_t)gptr;

    v4u g0;
    g0[0] = 0x1u;                                   // count=1, user descriptor
    g0[1] = lds_byte_addr;                          // LDS byte address
    g0[2] = (unsigned)(ga & 0xffffffffu);           // global_addr[31:0]
    g0[3] = (unsigned)((ga >> 32) & 0x1ffffffu)     // global_addr[56:32]
            | (2u << 30);                           // type = 2 ("image")

    v8u g1;
    g1[0] = (2u << 16)                              // data_size = 4 bytes
          | (1u << 20)                              // pad_enable
          | (4u << 22)                              // pad_interval: 32 DWORDs
          | (3u << 25);                             // pad_amount:   4 DWORDs
    g1[1] = (32u << 16);                            // tensor_dim0 = 32 (lo16)
    g1[2] = (16u << 16);                            // tensor_dim1 = 16 (lo16)
    g1[3] = (32u << 16);                            // tile_dim0 = 32
    g1[4] = 16u;                                    // tile_dim1 = 16, tile_dim2 = 0
    g1[5] = K_elems;                                // tensor_dim0_stride[31:0]
    g1[6] = 0u;                                     // stride hi16 / dim1_stride lo
    g1[7] = 0u;

    asm volatile("tensor_load_to_lds %0, %1" :: "s"(g0), "s"(g1) : "memory");
}

// ---------------------------------------------------------------------------
// Skinny GEMM: out[32,N] = A[32,K] * W[N,K]^T  (+ epilogue)
// 128 threads = 4 waves; each wave owns one 16-column tile, both 16-row halves.
// W tiles stream global->LDS via the Tensor Data Mover, double-buffered and
// software-pipelined (s_wait_tensorcnt).
// ---------------------------------------------------------------------------
#define GEMM_WAVES 4

__global__ __launch_bounds__(128)
void gemm32_kernel(const float* __restrict__ A, const float* __restrict__ W,
                   float* __restrict__ out, const float* __restrict__ extra,
                   int K, int N, int epi) {
    // [wave][buffer][row][36 floats] -- 36-float row stride = TDM pad result
    __shared__ __align__(16) float lds[GEMM_WAVES][2][16][36];

    const int wave = threadIdx.x >> 5;
    const int lane = threadIdx.x & 31;
    const int hi   = lane >> 4;        // 0: lanes 0-15, 1: lanes 16-31
    const int lp   = lane & 15;
    const int n0   = (blockIdx.x * GEMM_WAVES + wave) * 16;

    // LDS byte addresses of the two per-wave tile buffers (flat addr low 32
    // bits == LDS offset per ISA 10.2 aperture mapping)
    const unsigned ldsA0 = (unsigned)(uintptr_t)&lds[wave][0][0][0];
    const unsigned ldsA1 = (unsigned)(uintptr_t)&lds[wave][1][0][0];

    const float* wtile = W + (size_t)n0 * K;

    v8f c0 = {};   // rows 0..15
    v8f c1 = {};   // rows 16..31

    // prime the pipeline: chunk 0 -> buffer 0
    tdm_load_tile16x32(wtile, ldsA0, (unsigned)K);

    for (int k0 = 0; k0 < K; k0 += 32) {
        const int cur = (k0 >> 5) & 1;
        if (k0 + 32 < K) {
            // issue next chunk into the other buffer
            tdm_load_tile16x32(wtile + k0 + 32, cur ? ldsA0 : ldsA1, (unsigned)K);
            // keep next-next chunk warming L2 while TDM runs
            __builtin_prefetch(wtile + k0 + 64 + (size_t)lp * K + hi * 16, 0, 1);
            __builtin_amdgcn_s_wait_tensorcnt(1);   // current chunk landed
        } else {
            __builtin_amdgcn_s_wait_tensorcnt(0);
        }

#pragma unroll
        for (int t = 0; t < 8; ++t) {
            const int k = k0 + t * 4;
            // A fragment (16x4): lane lp holds row lp (c0) / lp+16 (c1), K pair at k + 2*hi
            v2f a0 = *(const v2f*)&A[(size_t)lp * K + k + 2 * hi];
            v2f a1 = *(const v2f*)&A[(size_t)(lp + 16) * K + k + 2 * hi];
            // B fragment (4x16): lane lp holds col n0+lp, K pair at 2*hi
            v2f b  = *(const v2f*)&lds[wave][cur][lp][t * 4 + 2 * hi];
            c0 = __builtin_amdgcn_wmma_f32_16x16x4_f32(false, a0, false, b, (short)0, c0, false, false);
            c1 = __builtin_amdgcn_wmma_f32_16x16x4_f32(false, a1, false, b, (short)0, c1, false, false);
        }
    }

    // D layout: VGPR v -> (M = v + 8*hi, N = n0 + lp)
    const int col = n0 + lp;
#pragma unroll
    for (int v = 0; v < 8; ++v) {
        int r0 = v + 8 * hi;
        int r1 = 16 + v + 8 * hi;
        out[(size_t)r0 * N + col] = epi_apply(c0[v], r0, col, N, epi, extra);
        out[(size_t)r1 * N + col] = epi_apply(c1[v], r1, col, N, epi, extra);
    }
}

// ---------------------------------------------------------------------------
// Segment average: altx[i, c1*32+c2] = mean_s x[c2, c1, i*64+s]
// ---------------------------------------------------------------------------
__global__ __launch_bounds__(256)
void avg_kernel(const float* __restrict__ x, float* __restrict__ altx) {
    int idx = blockIdx.x * 256 + threadIdx.x;       // 65536 outputs
    int i   = idx >> 11;
    int mm  = idx & 2047;
    int c1  = mm >> 5;
    int c2  = mm & 31;
    const float* p = x + ((size_t)(c2 * 64 + c1)) * TDIM + i * SEG;
    float s = 0.0f;
#pragma unroll 8
    for (int t = 0; t < SEG; ++t) s += p[t];
    altx[idx] = s * (1.0f / (float)SEG);
}

// ---------------------------------------------------------------------------
// LayerNorm over last dim (2048); one block per row. addres=1: out = LN(x)+x.
// ---------------------------------------------------------------------------
__global__ __launch_bounds__(256)
void ln_kernel(const float* __restrict__ in, const float* __restrict__ g,
               const float* __restrict__ b, float* __restrict__ out, int addres) {
    __shared__ float shs[256], shq[256];
    const int row = blockIdx.x;
    const int tid = threadIdx.x;
    const float* x = in + (size_t)row * MDIM;
    float s = 0.0f, q = 0.0f;
    for (int i = tid; i < MDIM; i += 256) { float v = x[i]; s += v; q += v * v; }
    shs[tid] = s; shq[tid] = q;
    __syncthreads();
    for (int off = 128; off > 0; off >>= 1) {
        if (tid < off) { shs[tid] += shs[tid + off]; shq[tid] += shq[tid + off]; }
        __syncthreads();
    }
    const float mean = shs[0] * (1.0f / MDIM);
    const float var  = shq[0] * (1.0f / MDIM) - mean * mean;
    const float rstd = rsqrtf(var + EPSL);
    for (int i = tid; i < MDIM; i += 256) {
        float v = x[i];
        float y = (v - mean) * rstd * g[i] + b[i];
        if (addres) y += v;
        out[(size_t)row * MDIM + i] = y;
    }
}

// ---------------------------------------------------------------------------
// rsa[i,h] = (1/sqrt(DH)) * sum_d q[i,h,d]*k[i,h,d]
// qkv layout: [32, 6144] rows; q at +0, k at +2048, v at +4096
// ---------------------------------------------------------------------------
__global__ __launch_bounds__(256)
void rsa_kernel(const float* __restrict__ qkv, float* __restrict__ rsa) {
    int t = blockIdx.x * 256 + threadIdx.x;         // 512 total
    if (t >= AVGF * HA) return;
    int i = t >> 4, h = t & 15;
    const float* q = qkv + (size_t)i * (3 * MDIM) + h * DH;
    const float* k = q + MDIM;
    float s = 0.0f;
#pragma unroll 4
    for (int d = 0; d < DH; ++d) s += q[d] * k[d];
    rsa[t] = s * 0.08838834764831845f;              // 1/sqrt(128)
}

// ---------------------------------------------------------------------------
// imv[i, h*128+d] = cumsum_i rsa[i,h] * v[i,h,d]
// ---------------------------------------------------------------------------
__global__ __launch_bounds__(256)
void scan_kernel(const float* __restrict__ qkv, const float* __restrict__ rsa,
                 float* __restrict__ imv) {
    int t = blockIdx.x * 256 + threadIdx.x;         // 2048 total: t = h*128+d
    int h = t >> 7;
    float acc = 0.0f;
    for (int i = 0; i < AVGF; ++i) {
        acc += rsa[i * HA + h] * qkv[(size_t)i * (3 * MDIM) + 2 * MDIM + t];
        imv[(size_t)i * MDIM + t] = acc;
    }
}

// ---------------------------------------------------------------------------
// Host-side orchestration
// ---------------------------------------------------------------------------
extern "C" void kernel_launch(void* const* d_in, const int* in_sizes, int n_in,
                              void* d_out, int out_size, void* d_ws, size_t ws_size,
                              hipStream_t stream) {
    const float* x      = (const float*)d_in[0];
    const float* weight = (const float*)d_in[1];
    const float* Wqkv   = (const float*)d_in[2];
    const float* Wo     = (const float*)d_in[3];
    const float* ln1_g  = (const float*)d_in[4];
    const float* ln1_b  = (const float*)d_in[5];
    const float* ln2_g  = (const float*)d_in[6];
    const float* ln2_b  = (const float*)d_in[7];
    const float* fc1_w  = (const float*)d_in[8];
    const float* fc1_b  = (const float*)d_in[9];
    const float* fc2_w  = (const float*)d_in[10];
    const float* fc2_b  = (const float*)d_in[11];

    float* ws   = (float*)d_ws;
    float* altx = ws;                    // 65536
    float* s    = ws + 65536;            // 65536
    float* y    = ws + 131072;           // 65536
    float* u    = ws + 196608;           // 65536
    float* hbuf = ws + 262144;           // 262144  (32 x 8192)
    float* qkv  = ws + 524288;           // 196608  (32 x 6144)
    float* rsa  = ws + 720896;           // 512
    float* imv  = ws + 721408;           // 65536

    // 1) segment-average -> altx [32, 2048]
    avg_kernel<<<256, 256, 0, stream>>>(x, altx);

    // 2) s = altx @ weight^T + sinusoidal bias
    gemm32_kernel<<<MDIM / 64, 128, 0, stream>>>(altx, weight, s, nullptr,
                                                 MDIM, MDIM, EPI_SINBIAS);

    for (int a = 0; a < 3; ++a) {
        // y = LN1(s)
        ln_kernel<<<AVGF, 256, 0, stream>>>(s, ln1_g, ln1_b, y, 0);

        // qkv[32, 6144] = y @ Wqkv[a]^T   (Wqkv[a] flattens to [6144, 2048])
        gemm32_kernel<<<(3 * MDIM) / 64, 128, 0, stream>>>(
            y, Wqkv + (size_t)a * 3 * MDIM * MDIM, qkv, nullptr,
            MDIM, 3 * MDIM, EPI_NONE);

        // per-token scalar attention + serial scan
        rsa_kernel<<<2, 256, 0, stream>>>(qkv, rsa);
        scan_kernel<<<8, 256, 0, stream>>>(qkv, rsa, imv);

        // u = imv @ Wo[a]^T + s
        gemm32_kernel<<<MDIM / 64, 128, 0, stream>>>(
            imv, Wo + (size_t)a * MDIM * MDIM, u, s, MDIM, MDIM, EPI_RES);

        // s = LN2(u) + u
        ln_kernel<<<AVGF, 256, 0, stream>>>(u, ln2_g, ln2_b, s, 1);

        // h = gelu(s @ fc1_w^T + fc1_b)   [32, 8192]
        gemm32_kernel<<<(4 * MDIM) / 64, 128, 0, stream>>>(
            s, fc1_w, hbuf, fc1_b, MDIM, 4 * MDIM, EPI_BIAS_GELU);

        // s = h @ fc2_w^T + fc2_b         [32, 2048]
        gemm32_kernel<<<MDIM / 64, 128, 0, stream>>>(
            hbuf, fc2_w, s, fc2_b, 4 * MDIM, MDIM, EPI_BIAS);
    }

    // s[32,2048] has the exact flat layout of out [32,64,32]
    hipMemcpyAsync(d_out, s, (size_t)AVGF * MDIM * sizeof(float),
                   hipMemcpyDeviceToDevice, stream);
}